// SaSSM_LoSh2D_22342419874200
// MI455X (gfx1250) — compile-verified
//
#include <hip/hip_runtime.h>

// ---------------- problem constants ----------------
#define BSZ 4
#define DM  96
#define DI  192
#define NS  16
#define RK  6
#define KD  4
#define LL  4096   // 64*64
#define XR  38     // dt_rank + 2*d_state
#define XRP 40     // padded row stride for x_dbl

typedef float v2f __attribute__((ext_vector_type(2)));
typedef float v8f __attribute__((ext_vector_type(8)));

struct cpx { float x, y; };

// CDNA5 native fp32 matrix op: D(16x16) = A(16x4) * B(4x16) + C, exact fp32.
__device__ __forceinline__ v8f wmma_f32(v2f a, v2f b, v8f c) {
  return __builtin_amdgcn_wmma_f32_16x16x4_f32(false, a, false, b, (short)0, c,
                                               false, false);
}

// ---------------- 1) in_proj GEMM: xz[16384,384] = x[16384,96] @ W[384,96]^T
__global__ void k_inproj(const float* __restrict__ X, const float* __restrict__ W,
                         float* __restrict__ xz) {
  int gt   = blockIdx.x * blockDim.x + threadIdx.x;
  int wid  = gt >> 5;
  int lane = threadIdx.x & 31;
  int half = lane >> 4;
  int lm   = lane & 15;
  int nt = wid % 24, mt = wid / 24;     // 24 N-tiles, 1024 M-tiles
  int m0 = mt * 16, n0 = nt * 16;
  v8f c;
#pragma unroll
  for (int j = 0; j < 8; ++j) c[j] = 0.f;
  int arow = m0 + lm;
  int bcol = n0 + lm;
  for (int k0 = 0; k0 < DM; k0 += 4) {
    int kb = k0 + half * 2;
    v2f a = *(const v2f*)(X + (size_t)arow * DM + kb);
    v2f b = *(const v2f*)(W + (size_t)bcol * DM + kb);
    c = wmma_f32(a, b, c);
  }
#pragma unroll
  for (int j = 0; j < 8; ++j) {
    int row = m0 + j + half * 8;
    xz[(size_t)row * (2 * DI) + n0 + lm] = c[j];
  }
}

// ---------------- 2) depthwise 3x3 conv + SiLU -> x_feat (B,DI,L) and (B,L,DI)
__global__ void k_conv(const float* __restrict__ xz, const float* __restrict__ cw,
                       const float* __restrict__ cb, float* __restrict__ xf,
                       float* __restrict__ xfT) {
  int gt = blockIdx.x * blockDim.x + threadIdx.x;
  if (gt >= BSZ * LL * DI) return;
  int c  = gt % DI;
  int pl = gt / DI;                 // b*L + l
  int b  = pl / LL;
  int l  = pl % LL;
  int h = l >> 6, w = l & 63;
  float acc = cb[c];
#pragma unroll
  for (int dh = -1; dh <= 1; ++dh) {
    int h2 = h + dh;
    if ((unsigned)h2 >= 64u) continue;
#pragma unroll
    for (int dw = -1; dw <= 1; ++dw) {
      int w2 = w + dw;
      if ((unsigned)w2 >= 64u) continue;
      acc += xz[((size_t)(b * LL + (h2 << 6) + w2)) * (2 * DI) + c] *
             cw[c * 9 + (dh + 1) * 3 + (dw + 1)];
    }
  }
  float s = acc / (1.f + __expf(-acc));    // SiLU
  xf[(size_t)(b * DI + c) * LL + l] = s;
  xfT[(size_t)pl * DI + c]          = s;
}

// ---------------- 3) phase-only FFT as 4 DFT passes (64x64 per (b,c) image)
// MODE 0: row forward (real in)     MODE 1: col forward + unit-normalize
// MODE 2: col inverse               MODE 3: row inverse, Re()/4096 -> reconT
template <int MODE>
__global__ void k_dft(const float* __restrict__ rin, const cpx* __restrict__ cin,
                      cpx* __restrict__ cout, float* __restrict__ rout) {
  __shared__ cpx S[4096];
  __shared__ cpx tw[64];
  int bc = blockIdx.x, tid = threadIdx.x;
  if (tid < 64) {
    float sn, cs;
    __sincosf(-6.283185307179586f * (float)tid / 64.f, &sn, &cs);
    tw[tid] = {cs, sn};
  }
  size_t base = (size_t)bc * 4096;
  for (int i = tid; i < 4096; i += 256) {
    if (MODE == 0) { S[i].x = rin[base + i]; S[i].y = 0.f; }
    else            S[i] = cin[base + i];
  }
  __syncthreads();
  for (int i = tid; i < 4096; i += 256) {
    int a = i >> 6, bcol = i & 63;
    float ax = 0.f, ay = 0.f;
    if (MODE == 0 || MODE == 3) {           // transform along rows
      for (int t = 0; t < 64; ++t) {
        cpx s = S[(a << 6) + t];
        cpx w = tw[(bcol * t) & 63];
        if (MODE == 0) { ax += s.x * w.x - s.y * w.y; ay += s.x * w.y + s.y * w.x; }
        else           { ax += s.x * w.x + s.y * w.y; ay += s.y * w.x - s.x * w.y; }
      }
    } else {                                 // transform along columns
      for (int t = 0; t < 64; ++t) {
        cpx s = S[(t << 6) + bcol];
        cpx w = tw[(a * t) & 63];
        if (MODE == 1) { ax += s.x * w.x - s.y * w.y; ay += s.x * w.y + s.y * w.x; }
        else           { ax += s.x * w.x + s.y * w.y; ay += s.y * w.x - s.x * w.y; }
      }
    }
    if (MODE == 1) {                         // exp(i*angle) == X/|X| (0 -> 1)
      float m2 = ax * ax + ay * ay;
      if (m2 > 0.f) { float r = rsqrtf(m2); ax *= r; ay *= r; }
      else          { ax = 1.f; ay = 0.f; }
    }
    if (MODE == 3) {
      int b = bc / DI, ch = bc % DI;
      rout[((size_t)(b * LL) + i) * DI + ch] = ax * (1.f / 4096.f);
    } else {
      cout[base + i] = {ax, ay};
    }
  }
}

// ---------------- 4) illum = channel mean
__global__ void k_illum(const float* __restrict__ xfT, float* __restrict__ illum) {
  int pix = blockIdx.x * blockDim.x + threadIdx.x;
  if (pix >= BSZ * LL) return;
  const float* r = xfT + (size_t)pix * DI;
  float s = 0.f;
  for (int i = 0; i < DI; ++i) s += r[i];
  illum[pix] = s * (1.f / (float)DI);
}

// ---------------- 5) phase-enhance MLP -> struct_map (B,4,L)
__global__ void k_pe(const float* __restrict__ reconT, const float* __restrict__ w1,
                     const float* __restrict__ b1, const float* __restrict__ w2,
                     const float* __restrict__ b2, float* __restrict__ smap) {
  __shared__ float rc[DI];
  __shared__ float ts[48];
  int pix = blockIdx.x;
  int b = pix >> 12, l = pix & 4095;
  int tid = threadIdx.x;
  for (int i = tid; i < DI; i += 64) rc[i] = reconT[(size_t)pix * DI + i];
  __syncthreads();
  if (tid < 48) {
    float acc = b1[tid];
    for (int c = 0; c < DI; ++c) acc += rc[c] * w1[tid * DI + c];
    ts[tid] = acc > 0.f ? acc : 0.2f * acc;      // leaky_relu(0.2)
  }
  __syncthreads();
  if (tid < 4) {
    float acc = b2[tid];
    for (int o = 0; o < 48; ++o) acc += ts[o] * w2[tid * 48 + o];
    smap[(size_t)(b * KD + tid) * LL + l] = 1.f / (1.f + __expf(-acc));
  }
}

// ---------------- 6) illum-gate conv+MLP+softmax -> ssa (B,L,4)
__global__ void k_ig(const float* __restrict__ illum, const float* __restrict__ w1,
                     const float* __restrict__ b1, const float* __restrict__ w2,
                     const float* __restrict__ b2, float* __restrict__ ssa4) {
  __shared__ float w1s[48 * 9], b1s[48], w2s[4 * 48], b2s[4];
  int tid = threadIdx.x;
  for (int i = tid; i < 432; i += 256) w1s[i] = w1[i];
  if (tid < 48) b1s[tid] = b1[tid];
  for (int i = tid; i < 192; i += 256) w2s[i] = w2[i];
  if (tid < 4) b2s[tid] = b2[tid];
  __syncthreads();
  int pix = blockIdx.x * 256 + tid;
  if (pix >= BSZ * LL) return;
  int b = pix >> 12, l = pix & 4095;
  int h = l >> 6, w = l & 63;
  float nb[9];
#pragma unroll
  for (int dh = -1; dh <= 1; ++dh)
#pragma unroll
    for (int dw = -1; dw <= 1; ++dw) {
      int h2 = h + dh, w2 = w + dw;
      nb[(dh + 1) * 3 + dw + 1] =
          ((unsigned)h2 < 64u && (unsigned)w2 < 64u) ? illum[b * LL + (h2 << 6) + w2] : 0.f;
    }
  float lg[4];
#pragma unroll
  for (int j = 0; j < 4; ++j) lg[j] = b2s[j];
  for (int o = 0; o < 48; ++o) {
    float g = b1s[o];
#pragma unroll
    for (int t = 0; t < 9; ++t) g += nb[t] * w1s[o * 9 + t];
    g = g > 0.f ? g : 0.2f * g;
#pragma unroll
    for (int j = 0; j < 4; ++j) lg[j] += g * w2s[j * 48 + o];
  }
  float mx = fmaxf(fmaxf(lg[0], lg[1]), fmaxf(lg[2], lg[3]));
  float e[4], se = 0.f;
#pragma unroll
  for (int j = 0; j < 4; ++j) { e[j] = __expf(lg[j] - mx); se += e[j]; }
  float inv = 1.f / se;
#pragma unroll
  for (int j = 0; j < 4; ++j) ssa4[(size_t)pix * 4 + j] = e[j] * inv;
}

// ---------------- 7) x_dbl[b,k,l,r] = (Wp[k] @ xs) * struct_map  (WMMA, gathered B)
__global__ void k_xdbl(const float* __restrict__ xfT, const float* __restrict__ Wp,
                       const int* __restrict__ sids, const float* __restrict__ smap,
                       float* __restrict__ xdbl) {
  int gt   = blockIdx.x * blockDim.x + threadIdx.x;
  int wid  = gt >> 5;
  int lane = threadIdx.x & 31;
  int half = lane >> 4;
  int lm   = lane & 15;
  int nt = wid & 255;               // 256 L-tiles
  int mt = (wid >> 8) % 3;          // 3 r-tiles (38 padded to 48)
  int bk = wid / 768;               // 16 (b,k) pairs
  int b = bk >> 2, k = bk & 3;
  int m0 = mt * 16, l0 = nt * 16;
  int lcol = l0 + lm;
  int pos  = sids[k * LL + lcol];
  int ar   = m0 + lm;
  int arc  = ar < XR ? ar : XR - 1; // clamp: rows >=38 discarded at store
  const float* wrow = Wp + (size_t)(k * XR + arc) * DI;
  const float* brow = xfT + (size_t)(b * LL + pos) * DI;
  v8f c;
#pragma unroll
  for (int j = 0; j < 8; ++j) c[j] = 0.f;
  for (int k0 = 0; k0 < DI; k0 += 4) {
    int kb = k0 + half * 2;
    v2f a  = *(const v2f*)(wrow + kb);
    v2f bb = *(const v2f*)(brow + kb);
    c = wmma_f32(a, bb, c);
  }
  float sm = smap[(size_t)bk * LL + lcol];
#pragma unroll
  for (int j = 0; j < 8; ++j) {
    int r = m0 + j + half * 8;
    if (r < XR) xdbl[((size_t)bk * LL + lcol) * XRP + r] = c[j] * sm;
  }
}

// ---------------- 8) selective scan (sequential over L); scatter == inverse perm
__global__ void k_scan(const float* __restrict__ xfT, const float* __restrict__ xdbl,
                       const int* __restrict__ sids, const float* __restrict__ dtw_g,
                       const float* __restrict__ dtb_g, const float* __restrict__ Alog_g,
                       const float* __restrict__ Ds_g, float* __restrict__ ysp) {
  __shared__ float sbuf[8 * XRP];
  __shared__ int   sid8[8];
  int bk = blockIdx.x;              // b*4 + k
  int b = bk >> 2, k = bk & 3;
  int d  = threadIdx.x;             // 0..191
  int kd = k * DI + d;
  float Areg[NS];
#pragma unroll
  for (int n = 0; n < NS; ++n) Areg[n] = -__expf(Alog_g[(size_t)kd * NS + n]);
  float dtw[RK];
#pragma unroll
  for (int r = 0; r < RK; ++r) dtw[r] = dtw_g[(size_t)kd * RK + r];
  float bias = dtb_g[kd];
  float Dd   = Ds_g[kd];
  float h[NS];
#pragma unroll
  for (int n = 0; n < NS; ++n) h[n] = 0.f;
  const float* xrow = xdbl + (size_t)bk * LL * XRP;
  const int*   sk   = sids + k * LL;
  const float* ub   = xfT + (size_t)b * LL * DI + d;
  float*       yb   = ysp + (size_t)bk * LL * DI + d;
  for (int l0 = 0; l0 < LL; l0 += 8) {
    if (l0 + 8 < LL) __builtin_prefetch(xrow + (size_t)(l0 + 8) * XRP, 0, 1);
    for (int i = d; i < 8 * XRP; i += DI) sbuf[i] = xrow[(size_t)l0 * XRP + i];
    if (d < 8) sid8[d] = sk[l0 + d];
    __syncthreads();
#pragma unroll
    for (int tt = 0; tt < 8; ++tt) {
      const float* row = sbuf + tt * XRP;
      int pos = sid8[tt];
      float u = ub[(size_t)pos * DI];
      float draw = bias;
#pragma unroll
      for (int r = 0; r < RK; ++r) draw += row[r] * dtw[r];
      float delta = (draw > 20.f) ? draw : __logf(1.f + __expf(draw));  // softplus
      float du = delta * u;
      float y = 0.f;
#pragma unroll
      for (int n = 0; n < NS; ++n) {
        h[n] = h[n] * __expf(delta * Areg[n]) + du * row[RK + n];
        y += h[n] * row[RK + NS + n];
      }
      yb[(size_t)pos * DI] = y + u * Dd;   // scatter to spatial = inverse perm
    }
    __syncthreads();
  }
}

// ---------------- 9) ssa-weighted K-sum + LayerNorm + silu(z) gate
__global__ void k_fuse(const float* __restrict__ ysp, const float* __restrict__ ssa4,
                       const float* __restrict__ xz, const float* __restrict__ lnw,
                       const float* __restrict__ lnb, float* __restrict__ yln) {
  __shared__ float2 red[DI];
  __shared__ float mu_s, rs_s;
  int pix = blockIdx.x;
  int d = threadIdx.x;
  int b = pix >> 12, l = pix & 4095;
  float val = 0.f;
#pragma unroll
  for (int kk = 0; kk < KD; ++kk)
    val += ysp[((size_t)((b * KD + kk) * LL) + l) * DI + d] * ssa4[(size_t)pix * 4 + kk];
  red[d] = make_float2(val, val * val);
  __syncthreads();
  for (int st = 96; st >= 3; st >>= 1) {
    if (d < st) { red[d].x += red[d + st].x; red[d].y += red[d + st].y; }
    __syncthreads();
  }
  if (d == 0) {
    float sx = red[0].x + red[1].x + red[2].x;
    float sy = red[0].y + red[1].y + red[2].y;
    float mu = sx / (float)DI;
    float var = sy / (float)DI - mu * mu;
    mu_s = mu;
    rs_s = rsqrtf(var + 1e-5f);
  }
  __syncthreads();
  float z  = xz[(size_t)pix * (2 * DI) + DI + d];
  float sz = z / (1.f + __expf(-z));
  yln[(size_t)pix * DI + d] = ((val - mu_s) * rs_s * lnw[d] + lnb[d]) * sz;
}

// ---------------- 10) out_proj GEMM: out[16384,96] = yln[16384,192] @ W[96,192]^T
__global__ void k_outproj(const float* __restrict__ A, const float* __restrict__ W,
                          float* __restrict__ out) {
  int gt   = blockIdx.x * blockDim.x + threadIdx.x;
  int wid  = gt >> 5;
  int lane = threadIdx.x & 31;
  int half = lane >> 4;
  int lm   = lane & 15;
  int nt = wid % 6, mt = wid / 6;
  int m0 = mt * 16, n0 = nt * 16;
  v8f c;
#pragma unroll
  for (int j = 0; j < 8; ++j) c[j] = 0.f;
  int arow = m0 + lm;
  int bcol = n0 + lm;
  for (int k0 = 0; k0 < DI; k0 += 4) {
    int kb = k0 + half * 2;
    v2f a = *(const v2f*)(A + (size_t)arow * DI + kb);
    v2f b = *(const v2f*)(W + (size_t)bcol * DI + kb);
    c = wmma_f32(a, b, c);
  }
#pragma unroll
  for (int j = 0; j < 8; ++j)
    out[(size_t)(m0 + j + half * 8) * DM + n0 + lm] = c[j];
}

// ---------------- launch ----------------
extern "C" void kernel_launch(void* const* d_in, const int* in_sizes, int n_in,
                              void* d_out, int out_size, void* d_ws, size_t ws_size,
                              hipStream_t stream) {
  (void)in_sizes; (void)n_in; (void)out_size; (void)ws_size;
  const float* x        = (const float*)d_in[0];
  const float* in_w     = (const float*)d_in[1];
  const float* conv_w   = (const float*)d_in[2];
  const float* conv_b   = (const float*)d_in[3];
  const float* pe_w1    = (const float*)d_in[4];
  const float* pe_b1    = (const float*)d_in[5];
  const float* pe_w2    = (const float*)d_in[6];
  const float* pe_b2    = (const float*)d_in[7];
  const float* ig_w1    = (const float*)d_in[8];
  const float* ig_b1    = (const float*)d_in[9];
  const float* ig_w2    = (const float*)d_in[10];
  const float* ig_b2    = (const float*)d_in[11];
  const float* xproj_w  = (const float*)d_in[12];
  const float* dt_w     = (const float*)d_in[13];
  const float* dt_b     = (const float*)d_in[14];
  const float* A_logs   = (const float*)d_in[15];
  const float* Ds       = (const float*)d_in[16];
  const float* ln_w     = (const float*)d_in[17];
  const float* ln_b     = (const float*)d_in[18];
  const float* out_w    = (const float*)d_in[19];
  const int*   scan_ids = (const int*)d_in[20];
  float* out = (float*)d_out;

  char* w = (char*)d_ws;
  float* xz     = (float*)(w + 0);            // 25,165,824 B
  float* xf     = (float*)(w + 25165824);     // 12,582,912 B (B,DI,L)
  float* xfT    = (float*)(w + 37748736);     // 12,582,912 B (B,L,DI)
  float* reconT = (float*)(w + 50331648);     // 12,582,912 B (aliased as yln later)
  float* smap   = (float*)(w + 62914560);     //    262,144 B
  float* illum  = (float*)(w + 63176704);     //     65,536 B
  float* ssa4   = (float*)(w + 63242240);     //    262,144 B
  float* xdbl   = (float*)(w + 63504384);     // 10,485,760 B
  float* ysp    = (float*)(w + 73990144);     // 50,331,648 B
  cpx*   c1     = (cpx*)(w + 73990144);       // FFT ping (aliases ysp; FFT precedes scan)
  cpx*   c2     = (cpx*)(w + 73990144 + 25165824);
  float* yln    = reconT;                     // recon dead after k_pe

  k_inproj<<<3072, 256, 0, stream>>>(x, in_w, xz);
  k_conv  <<<12288, 256, 0, stream>>>(xz, conv_w, conv_b, xf, xfT);
  k_dft<0><<<768, 256, 0, stream>>>(xf, nullptr, c1, nullptr);
  k_dft<1><<<768, 256, 0, stream>>>(nullptr, c1, c2, nullptr);
  k_dft<2><<<768, 256, 0, stream>>>(nullptr, c2, c1, nullptr);
  k_dft<3><<<768, 256, 0, stream>>>(nullptr, c1, nullptr, reconT);
  k_illum <<<64, 256, 0, stream>>>(xfT, illum);
  k_pe    <<<16384, 64, 0, stream>>>(reconT, pe_w1, pe_b1, pe_w2, pe_b2, smap);
  k_ig    <<<64, 256, 0, stream>>>(illum, ig_w1, ig_b1, ig_w2, ig_b2, ssa4);
  k_xdbl  <<<1536, 256, 0, stream>>>(xfT, xproj_w, scan_ids, smap, xdbl);
  k_scan  <<<16, 192, 0, stream>>>(xfT, xdbl, scan_ids, dt_w, dt_b, A_logs, Ds, ysp);
  k_fuse  <<<16384, 192, 0, stream>>>(ysp, ssa4, xz, ln_w, ln_b, yln);
  k_outproj<<<768, 256, 0, stream>>>(yln, out_w, out);
}